// Model_75170517615174
// MI455X (gfx1250) — compile-verified
//
#include <hip/hip_runtime.h>

// MI455X / gfx1250 implementation.
// Memory-bound iterative masked-symmetric soft-threshold solver.
// A_hat kept in-place in d_out; row reductions done with V_WMMA_F32_16X16X4_F32
// (A_tile x Ones accumulation, exact f32) + global f32 atomics.

#define LDIM 1024
#define BDIM 4
#define TS 32
#define NT (LDIM / TS)                 // 32 tiles per dim
#define NPAIR ((NT * (NT + 1)) / 2)    // 528 tile pairs (TI <= TJ)

typedef float v2f __attribute__((ext_vector_type(2)));
typedef float v8f __attribute__((ext_vector_type(8)));

__global__ __launch_bounds__(256) void zero_f32_kernel(float* __restrict__ p, int n) {
  int i = blockIdx.x * blockDim.x + threadIdx.x;
  if (i < n) p[i] = 0.0f;
}

// Fold rowsums into Lm, produce c[i] = Lm[i]*sign(row[i]-1), re-zero row buffer.
__global__ __launch_bounds__(256) void lm_update_kernel(
    float* __restrict__ rowbuf, float* __restrict__ Lm, float* __restrict__ cvec,
    const float* __restrict__ w_p, const float* __restrict__ belt_p,
    const float* __restrict__ lr_belt_p, float t, int init) {
  int i = blockIdx.x * blockDim.x + threadIdx.x;   // i < BDIM*LDIM (exact multiple)
  float rv = rowbuf[i] - 1.0f;
  rowbuf[i] = 0.0f;                                 // ready for next step's atomics
  float coeff = init ? w_p[0] : (belt_p[0] * powf(lr_belt_p[0], t));
  float lm = (init ? 0.0f : Lm[i]) + coeff * fmaxf(rv, 0.0f);
  Lm[i] = lm;
  float sg = (rv > 0.0f) ? 1.0f : ((rv < 0.0f) ? -1.0f : 0.0f);
  cvec[i] = lm * sg;
}

// One optimization step, fused: A_hat update (in-place safe: each block owns the
// (I,J)+(J,I) tile pair), plus rowsum(A) partials via WMMA reductions.
// init=1: at=0 -> A_hat(dst) = src exactly (x in [0,1)), rowsums of A0 produced.
__global__ __launch_bounds__(256) void step_kernel(
    const float* __restrict__ src, float* __restrict__ dst,
    const float* __restrict__ x, const float* __restrict__ Mmask,
    const float* __restrict__ rho, const float* __restrict__ cvec,
    float* __restrict__ rowbuf,
    const float* __restrict__ s_p, const float* __restrict__ alpha_p,
    const float* __restrict__ lr_alpha_p, float t, int init) {
  __shared__ float tA[TS][TS + 1];   // transposed staging (A_hat, then h2 out)
  __shared__ float tX[TS][TS + 1];   // transposed x
  __shared__ float tR[TS][TS + 1];   // transposed rho_p
  __shared__ float aT[TS][TS + 1];   // a-tile for WMMA reductions

  int bidx = blockIdx.x;
  int b = bidx / NPAIR;
  int p = bidx % NPAIR;
  int ti = 0, rl = NT;
  while (p >= rl) { p -= rl; rl--; ti++; }
  int tj = ti + p;
  const int I0 = ti * TS, J0 = tj * TS;
  const float wscale = (ti == tj) ? 0.5f : 1.0f;  // diag tiles counted twice

  const int tid = threadIdx.x;
  const int cc = tid & 31;     // column within tile
  const int r8 = tid >> 5;     // row group (0..7), rows r8 + 8*it

  const float s = s_p[0];
  const float at = init ? 0.0f : (alpha_p[0] * powf(lr_alpha_p[0], t));

  const size_t base = (size_t)b * LDIM * LDIM;
  const float* __restrict__ srcB = src + base;
  const float* __restrict__ xB = x + base;
  const float* __restrict__ MB = Mmask + base;
  float* __restrict__ dstB = dst + base;
  const float* __restrict__ cB = cvec + b * LDIM;

  float h1o[4], h2o[4], xij[4], xji[4], rij[4], rji[4], mm[4], civ[4];
  const float cj = cB[J0 + cc];

  // Direct (coalesced) loads of tile1 = (rows I0+r, cols J0+cc)
#pragma unroll
  for (int it = 0; it < 4; ++it) {
    int r = r8 + 8 * it;
    size_t o = (size_t)(I0 + r) * LDIM + (J0 + cc);
    h1o[it] = srcB[o];
    xij[it] = xB[o];
    mm[it]  = MB[o];
    rij[it] = rho[(size_t)(I0 + r) * LDIM + (J0 + cc)];
    civ[it] = cB[I0 + r];
  }
  // Coalesced loads of tile2 = (rows J0+r, cols I0+cc) staged to LDS
#pragma unroll
  for (int it = 0; it < 4; ++it) {
    int r = r8 + 8 * it;
    size_t o = (size_t)(J0 + r) * LDIM + (I0 + cc);
    tA[r][cc] = srcB[o];
    tX[r][cc] = xB[o];
    tR[r][cc] = rho[(size_t)(J0 + r) * LDIM + (I0 + cc)];
  }
  // All global loads must land before any thread stores (diag tiles, in-place).
  asm volatile("s_wait_loadcnt 0x0" ::: "memory");
  __syncthreads();
#pragma unroll
  for (int it = 0; it < 4; ++it) {
    int r = r8 + 8 * it;
    h2o[it] = tA[cc][r];   // A_hat[j][i]
    xji[it] = tX[cc][r];   // x[j][i]
    rji[it] = tR[cc][r];   // rho[j][i]
  }
  __syncthreads();         // tA reused below for h2 store staging

#pragma unroll
  for (int it = 0; it < 4; ++it) {
    int r = r8 + 8 * it;
    // symmetric multiplier pieces, shared between (i,j) and (j,i)
    float usum = (xij[it] + xji[it]) * 0.5f - s;     // Us[i][j]
    float g = usum - civ[it] - cj;                   // (G + G^T)[i][j]
    float mv = mm[it];                               // M symmetric
    float h1 = h1o[it] + at * h1o[it] * mv * g;
    h1 = fmaxf(fabsf(h1) - rij[it] * at, 0.0f);
    h1 = 1.0f - fmaxf(1.0f - h1, 0.0f);
    float h2 = h2o[it] + at * h2o[it] * mv * g;
    h2 = fmaxf(fabsf(h2) - rji[it] * at, 0.0f);
    h2 = 1.0f - fmaxf(1.0f - h2, 0.0f);
    float a = (h1 + h2) * 0.5f * mv;                 // A[i][j] == A[j][i]
    dstB[(size_t)(I0 + r) * LDIM + (J0 + cc)] = h1;  // tile1 store, coalesced
    tA[cc][r] = h2;                                  // stage tile2 transposed
    aT[r][cc] = a * wscale;                          // reduction tile
  }
  __syncthreads();
  // Coalesced store of tile2 (for diag: duplicate store of identical values)
#pragma unroll
  for (int it = 0; it < 4; ++it) {
    int r = r8 + 8 * it;
    dstB[(size_t)(J0 + r) * LDIM + (I0 + cc)] = tA[r][cc];
  }

  // ---- WMMA reduction: row/col sums of the 32x32 a-tile ----
  // 8 waves: mode 0 -> rowsums (rows I0+...), mode 1 -> colsums (rows J0+...),
  // each wave owns one 16x16 subtile; D = Asub x Ones via 4x WMMA f32 16x16x4.
  const int wv = tid >> 5;
  const int lane = tid & 31;
  const int mode = wv >> 2;
  const int sub = wv & 3;
  const int si = (sub >> 1) * 16;
  const int sj = (sub & 1) * 16;
  const int ml = lane & 15;
  const int kh = (lane >> 4) << 1;   // K offset per lane half (0 or 2)

  v8f acc = (v8f){0.f, 0.f, 0.f, 0.f, 0.f, 0.f, 0.f, 0.f};
  v2f bv;  bv.x = 1.0f;  bv.y = 1.0f;   // ones B-matrix (layout-independent)
#pragma unroll
  for (int kb = 0; kb < 4; ++kb) {
    int k0 = 4 * kb + kh;
    v2f av;
    if (mode == 0) {                 // rowsum: A[m][k] = aT[si+m][sj+k]
      av.x = aT[si + ml][sj + k0];
      av.y = aT[si + ml][sj + k0 + 1];
    } else {                         // colsum: A'[n][k] = aT[si+k][sj+n]
      av.x = aT[si + k0][sj + ml];
      av.y = aT[si + k0 + 1][sj + ml];
    }
    acc = __builtin_amdgcn_wmma_f32_16x16x4_f32(
        false, av, false, bv, (short)0, acc, false, false);
  }
  // D[m][n] = sum over K, replicated over N. Lanes 0/16 hold rows 0-7 / 8-15.
  const int outBase = (mode == 0) ? (I0 + si) : (J0 + sj);
  float* __restrict__ rb = rowbuf + b * LDIM;
  if (lane == 0) {
#pragma unroll
    for (int q = 0; q < 8; ++q) atomicAdd(&rb[outBase + q], acc[q]);
  } else if (lane == 16) {
#pragma unroll
    for (int q = 0; q < 8; ++q) atomicAdd(&rb[outBase + 8 + q], acc[q]);
  }
}

// Final output: A = (A_hat + A_hat^T) * 0.5 * M, in-place over A_hat (d_out).
__global__ __launch_bounds__(256) void final_kernel(
    float* __restrict__ Ad, const float* __restrict__ Mmask) {
  __shared__ float tH[TS][TS + 1];
  int bidx = blockIdx.x;
  int b = bidx / NPAIR;
  int p = bidx % NPAIR;
  int ti = 0, rl = NT;
  while (p >= rl) { p -= rl; rl--; ti++; }
  int tj = ti + p;
  const int I0 = ti * TS, J0 = tj * TS;
  const int tid = threadIdx.x;
  const int cc = tid & 31;
  const int r8 = tid >> 5;
  const size_t base = (size_t)b * LDIM * LDIM;
  float* __restrict__ AB = Ad + base;
  const float* __restrict__ MB = Mmask + base;

  float h1[4], mm[4];
#pragma unroll
  for (int it = 0; it < 4; ++it) {
    int r = r8 + 8 * it;
    size_t o = (size_t)(I0 + r) * LDIM + (J0 + cc);
    h1[it] = AB[o];
    mm[it] = MB[o];
  }
#pragma unroll
  for (int it = 0; it < 4; ++it) {
    int r = r8 + 8 * it;
    tH[r][cc] = AB[(size_t)(J0 + r) * LDIM + (I0 + cc)];
  }
  asm volatile("s_wait_loadcnt 0x0" ::: "memory");
  __syncthreads();
  float h2[4];
#pragma unroll
  for (int it = 0; it < 4; ++it) h2[it] = tH[cc][r8 + 8 * it];
  __syncthreads();
#pragma unroll
  for (int it = 0; it < 4; ++it) {
    int r = r8 + 8 * it;
    float a = (h1[it] + h2[it]) * 0.5f * mm[it];
    AB[(size_t)(I0 + r) * LDIM + (J0 + cc)] = a;
    tH[cc][r] = a;                       // A[j][i] = A[i][j] (M symmetric)
  }
  __syncthreads();
#pragma unroll
  for (int it = 0; it < 4; ++it) {
    int r = r8 + 8 * it;
    AB[(size_t)(J0 + r) * LDIM + (I0 + cc)] = tH[r][cc];
  }
}

extern "C" void kernel_launch(void* const* d_in, const int* in_sizes, int n_in,
                              void* d_out, int out_size, void* d_ws, size_t ws_size,
                              hipStream_t stream) {
  (void)in_sizes; (void)n_in; (void)out_size; (void)ws_size;
  const float* x      = (const float*)d_in[0];
  const float* M      = (const float*)d_in[1];
  const float* s_p    = (const float*)d_in[2];
  const float* w_p    = (const float*)d_in[3];
  const float* rho    = (const float*)d_in[4];
  const float* alpha  = (const float*)d_in[5];
  const float* belt   = (const float*)d_in[6];
  const float* lr_a   = (const float*)d_in[7];
  const float* lr_b   = (const float*)d_in[8];
  const int STEPS = 16;  // reference steps (python scalar, fixed)

  float* Ahat = (float*)d_out;            // A_hat lives in d_out during iters
  float* row  = (float*)d_ws;             // [B*L] rowsum accumulator
  float* Lm   = row + BDIM * LDIM;        // [B*L]
  float* cv   = Lm + BDIM * LDIM;         // [B*L] c = Lm*sign(row-1)

  const dim3 grid(BDIM * NPAIR), blk(256);
  const dim3 vgrid((BDIM * LDIM) / 256), vblk(256);

  zero_f32_kernel<<<vgrid, vblk, 0, stream>>>(row, BDIM * LDIM);
  // init: A_hat = x (at==0 makes update identity on x in [0,1)), rowsum(A0)
  step_kernel<<<grid, blk, 0, stream>>>(x, Ahat, x, M, rho, cv, row,
                                        s_p, alpha, lr_a, 0.0f, 1);
  lm_update_kernel<<<vgrid, vblk, 0, stream>>>(row, Lm, cv, w_p, belt, lr_b,
                                               0.0f, 1);
  for (int t = 0; t < STEPS; ++t) {
    step_kernel<<<grid, blk, 0, stream>>>(Ahat, Ahat, x, M, rho, cv, row,
                                          s_p, alpha, lr_a, (float)t, 0);
    if (t < STEPS - 1)
      lm_update_kernel<<<vgrid, vblk, 0, stream>>>(row, Lm, cv, w_p, belt, lr_b,
                                                   (float)t, 0);
  }
  final_kernel<<<grid, blk, 0, stream>>>(Ahat, M);
}